// GATEncoder_31336081392101
// MI455X (gfx1250) — compile-verified
//
#include <hip/hip_runtime.h>
#include <hip/hip_bf16.h>
#include <math.h>

typedef float v2f __attribute__((ext_vector_type(2)));
typedef float v8f __attribute__((ext_vector_type(8)));

#define NEG_SLOPE 0.2f
#define BN_EPS 1e-5f

// -------------------- helpers --------------------

__device__ __forceinline__ void atomicMaxF(float* addr, float val) {
  // Sign-split trick: correct total order for IEEE floats.
  if (val >= 0.0f) atomicMax((int*)addr, __float_as_int(val));
  else             atomicMin((unsigned int*)addr, __float_as_uint(val));
}

__global__ void k_fill(float* __restrict__ p, float v, long n) {
  long i = (long)blockIdx.x * blockDim.x + threadIdx.x;
  if (i < n) p[i] = v;
}

// -------------------- layer-1 GEMM (K=5, scalar: negligible FLOPs) --------------------

__global__ void k_gemm_in5(const float* __restrict__ x, const float* __restrict__ W,
                           float* __restrict__ H, int N) {
  long i = (long)blockIdx.x * blockDim.x + threadIdx.x;
  if (i >= (long)N * 128) return;
  int n = (int)(i >> 7), o = (int)(i & 127);
  const float* xr = x + (long)n * 5;
  float s = 0.f;
#pragma unroll
  for (int k = 0; k < 5; ++k) s = fmaf(xr[k], W[k * 128 + o], s);
  H[i] = s;
}

// -------------------- fp32 WMMA GEMM: A[M,128] x B[128,NOUT] -> C[M,NOUT] --------------------
// One 16x16 output tile per wave via V_WMMA_F32_16X16X4_F32 (K-loop of 32 steps).
// A layout (16x4 fp32): lanes 0-15 hold K=0,1 ; lanes 16-31 hold K=2,3 (M = lane&15).
// B layout (4x16 fp32): symmetric (N = lane&15, K split by lane half).
// C/D layout: VGPR r -> M = r (+8 for upper lane half), N = lane&15.

template <int NOUT>
__global__ __launch_bounds__(2 * NOUT)
void k_gemm_wmma(const float* __restrict__ A, const float* __restrict__ B,
                 float* __restrict__ C, int M) {
  __shared__ float Bt[NOUT * 128];              // B transposed: Bt[n][k]
  const int nthr = 2 * NOUT;
  int tid = threadIdx.x;
  for (int i = tid; i < NOUT * 128; i += nthr) {
    int k = i & 127, n = i >> 7;
    Bt[i] = B[k * NOUT + n];
  }
  __syncthreads();

  int wave = tid >> 5, lane = tid & 31;
  int m0 = blockIdx.x * 16;
  int n0 = wave * 16;
  int koff = (lane >> 4) * 2;                   // 0 or 2
  int mrow = m0 + (lane & 15);
  if (mrow >= M) mrow = M - 1;                  // clamp (stores are guarded)
  const float* Arow = A + (size_t)mrow * 128;
  const float* Bl   = &Bt[(size_t)(n0 + (lane & 15)) * 128];

  v8f acc = {};
#pragma unroll
  for (int kk = 0; kk < 128; kk += 4) {
    v2f a = *(const v2f*)(Arow + kk + koff);    // global_load_b64
    v2f b = *(const v2f*)(Bl   + kk + koff);    // ds_load_b64
    acc = __builtin_amdgcn_wmma_f32_16x16x4_f32(
        /*neg_a=*/false, a, /*neg_b=*/false, b,
        /*c_mod=*/(short)0, acc, /*reuse_a=*/false, /*reuse_b=*/false);
  }

  int ncol  = n0 + (lane & 15);
  int mbase = m0 + ((lane >> 4) << 3);
#pragma unroll
  for (int r = 0; r < 8; ++r) {
    int mr = mbase + r;
    if (mr < M) C[(size_t)mr * NOUT + ncol] = acc[r];
  }
}

// -------------------- attention scores a_s, a_d --------------------

__global__ void k_attn(const float* __restrict__ Hf, const float* __restrict__ atts,
                       const float* __restrict__ attd, float* __restrict__ aS,
                       float* __restrict__ aD, int N, int Hh, int hShift, int C) {
  long i = (long)blockIdx.x * blockDim.x + threadIdx.x;
  if (i >= ((long)N << hShift)) return;
  int n = (int)(i >> hShift), h = (int)(i & (Hh - 1));
  const float* hp = Hf + (size_t)n * (Hh * C) + (size_t)h * C;
  float ss = 0.f, sd = 0.f;
  for (int c = 0; c < C; ++c) {
    float v = hp[c];
    ss = fmaf(v, atts[h * C + c], ss);
    sd = fmaf(v, attd[h * C + c], sd);
  }
  aS[i] = ss;
  aD[i] = sd;
}

// -------------------- edge passes (segment softmax + aggregation) --------------------

__global__ void k_edge_logit(const int* __restrict__ ei, int Eorig, int Etot,
                             const float* __restrict__ aS, const float* __restrict__ aD,
                             float* __restrict__ pE, float* __restrict__ mM,
                             int Hh, int hShift) {
  long i = (long)blockIdx.x * blockDim.x + threadIdx.x;
  if (i >= ((long)Etot << hShift)) return;
  int e = (int)(i >> hShift), h = (int)(i & (Hh - 1));
  int s, d;
  if (e < Eorig) { s = ei[e]; d = ei[Eorig + e]; } else { s = d = e - Eorig; }
  float l = aS[(long)s * Hh + h] + aD[(long)d * Hh + h];
  l = (l >= 0.f) ? l : NEG_SLOPE * l;
  pE[i] = l;
  atomicMaxF(&mM[(long)d * Hh + h], l);
}

__global__ void k_edge_exp(const int* __restrict__ ei, int Eorig, int Etot,
                           float* __restrict__ pE, const float* __restrict__ mM,
                           float* __restrict__ dn, int Hh, int hShift) {
  long i = (long)blockIdx.x * blockDim.x + threadIdx.x;
  if (i >= ((long)Etot << hShift)) return;
  int e = (int)(i >> hShift), h = (int)(i & (Hh - 1));
  int d;
  if (e < Eorig) d = ei[Eorig + e]; else d = e - Eorig;
  float p = expf(pE[i] - mM[(long)d * Hh + h]);
  pE[i] = p;
  atomicAdd(&dn[(long)d * Hh + h], p);
}

__global__ void k_edge_agg(const int* __restrict__ ei, int Eorig, int Etot,
                           const float* __restrict__ HW, const float* __restrict__ pE,
                           const float* __restrict__ dn, float* __restrict__ agg,
                           int Hh, int CH, int tpeShift, int cShift) {
  long i = (long)blockIdx.x * blockDim.x + threadIdx.x;
  if (i >= ((long)Etot << tpeShift)) return;
  int e = (int)(i >> tpeShift);
  int t = (int)(i & ((1 << tpeShift) - 1));
  int c0 = t << 2;                 // 4 channels per lane (float4)
  int h  = c0 >> cShift;           // owning head
  int s, d;
  if (e < Eorig) { s = ei[e]; d = ei[Eorig + e]; } else { s = d = e - Eorig; }
  float alpha = pE[(long)e * Hh + h] / (dn[(long)d * Hh + h] + 1e-16f);
  const float4 hv = *(const float4*)(HW + (size_t)s * CH + c0);
  float* op = agg + (size_t)d * CH + c0;
  atomicAdd(op + 0, hv.x * alpha);
  atomicAdd(op + 1, hv.y * alpha);
  atomicAdd(op + 2, hv.z * alpha);
  atomicAdd(op + 3, hv.w * alpha);
}

// -------------------- BatchNorm (eval on batch stats, ddof=0) + ELU --------------------

__global__ void k_bn_stats(const float* __restrict__ agg, const float* __restrict__ bias,
                           float* __restrict__ sums, int N, int CH) {
  int c = threadIdx.x;  // blockDim.x == CH
  float b = bias[c];
  float s = 0.f, s2 = 0.f;
  for (int n = blockIdx.x; n < N; n += gridDim.x) {
    float v = agg[(size_t)n * CH + c] + b;
    s += v;
    s2 = fmaf(v, v, s2);
  }
  atomicAdd(&sums[c], s);
  atomicAdd(&sums[CH + c], s2);
}

__global__ void k_bn_elu(const float* __restrict__ agg, const float* __restrict__ bias,
                         const float* __restrict__ sums, const float* __restrict__ g,
                         const float* __restrict__ be, float* __restrict__ out,
                         int N, int CH) {
  long i = (long)blockIdx.x * blockDim.x + threadIdx.x;
  if (i >= (long)N * CH) return;
  int c = (int)(i & (CH - 1));
  float invN = 1.0f / (float)N;
  float mu  = sums[c] * invN;
  float var = sums[CH + c] * invN - mu * mu;
  float v = (agg[i] + bias[c] - mu) * rsqrtf(var + BN_EPS) * g[c] + be[c];
  out[i] = (v > 0.f) ? v : expm1f(v);
}

__global__ void k_elu_bias(const float* __restrict__ agg, const float* __restrict__ b,
                           float* __restrict__ out, long n, int C) {
  long i = (long)blockIdx.x * blockDim.x + threadIdx.x;
  if (i >= n) return;
  int c = (int)(i & (C - 1));
  float v = agg[i] + b[c];
  out[i] = (v > 0.f) ? v : expm1f(v);
}

// -------------------- graph pooling (mean || max) --------------------

__global__ void k_pool_init(float* __restrict__ out, float* __restrict__ counts, int G) {
  int i = blockIdx.x * blockDim.x + threadIdx.x;
  if (i < G * 128) {
    int c = i & 127;
    out[i] = (c < 64) ? 0.f : -INFINITY;
  }
  if (i < G) counts[i] = 0.f;
}

__global__ void k_pool_count(const int* __restrict__ batch, float* __restrict__ counts, int N) {
  int i = blockIdx.x * blockDim.x + threadIdx.x;
  if (i >= N) return;
  atomicAdd(&counts[batch[i]], 1.0f);
}

__global__ void k_pool_accum(const float* __restrict__ h, const int* __restrict__ batch,
                             float* __restrict__ out, int N) {
  long i = (long)blockIdx.x * blockDim.x + threadIdx.x;
  if (i >= (long)N * 64) return;
  int n = (int)(i >> 6), c = (int)(i & 63);
  int g = batch[n];
  float v = h[i];
  atomicAdd(&out[g * 128 + c], v);
  atomicMaxF(&out[g * 128 + 64 + c], v);
}

__global__ void k_pool_final(float* __restrict__ out, const float* __restrict__ counts, int G) {
  int i = blockIdx.x * blockDim.x + threadIdx.x;
  if (i >= G * 128) return;
  int g = i >> 7, c = i & 127;
  float cnt = counts[g];
  if (c < 64)        out[i] = out[i] / fmaxf(cnt, 1.0f);
  else if (cnt <= 0) out[i] = 0.0f;
}

// -------------------- host driver --------------------

static inline unsigned cdiv(long n, int b) { return (unsigned)((n + b - 1) / b); }

static void run_edges(hipStream_t stream, const int* ei, int Eorig, int Etot, int N,
                      const float* HW, const float* aS, const float* aD,
                      float* mM, float* dn, float* pE, float* agg,
                      int Hh, int hShift, int CH, int cShift, int tpeShift) {
  const int B = 256;
  k_fill<<<cdiv((long)N * Hh, B), B, 0, stream>>>(mM, -INFINITY, (long)N * Hh);
  k_fill<<<cdiv((long)N * Hh, B), B, 0, stream>>>(dn, 0.f, (long)N * Hh);
  k_fill<<<cdiv((long)N * CH, B), B, 0, stream>>>(agg, 0.f, (long)N * CH);
  k_edge_logit<<<cdiv((long)Etot << hShift, B), B, 0, stream>>>(ei, Eorig, Etot, aS, aD, pE, mM, Hh, hShift);
  k_edge_exp<<<cdiv((long)Etot << hShift, B), B, 0, stream>>>(ei, Eorig, Etot, pE, mM, dn, Hh, hShift);
  k_edge_agg<<<cdiv((long)Etot << tpeShift, B), B, 0, stream>>>(ei, Eorig, Etot, HW, pE, dn, agg, Hh, CH, tpeShift, cShift);
}

extern "C" void kernel_launch(void* const* d_in, const int* in_sizes, int n_in,
                              void* d_out, int out_size, void* d_ws, size_t ws_size,
                              hipStream_t stream) {
  const float* x    = (const float*)d_in[0];
  const int*   ei   = (const int*)d_in[1];
  const int*   batch= (const int*)d_in[2];
  const float* W1   = (const float*)d_in[3];
  const float* a1s  = (const float*)d_in[4];
  const float* a1d  = (const float*)d_in[5];
  const float* b1   = (const float*)d_in[6];
  const float* g1   = (const float*)d_in[7];
  const float* be1  = (const float*)d_in[8];
  const float* W2   = (const float*)d_in[9];
  const float* a2s  = (const float*)d_in[10];
  const float* a2d  = (const float*)d_in[11];
  const float* b2   = (const float*)d_in[12];
  const float* g2   = (const float*)d_in[13];
  const float* be2  = (const float*)d_in[14];
  const float* W3   = (const float*)d_in[15];
  const float* a3s  = (const float*)d_in[16];
  const float* a3d  = (const float*)d_in[17];
  const float* b3   = (const float*)d_in[18];
  float* out = (float*)d_out;

  const int N     = in_sizes[0] / 5;
  const int Eorig = in_sizes[1] / 2;
  const int Etot  = Eorig + N;
  const int G     = 64;

  float* ws = (float*)d_ws;
  size_t o = 0;
  float* bufH   = ws + o; o += (size_t)N * 128;   // layer input / final h3
  float* bufHW  = ws + o; o += (size_t)N * 128;   // h = input @ W (pre-aggregation)
  float* bufAgg = ws + o; o += (size_t)N * 128;   // aggregated output
  float* aS     = ws + o; o += (size_t)N * 4;
  float* aD     = ws + o; o += (size_t)N * 4;
  float* mM     = ws + o; o += (size_t)N * 4;
  float* dn     = ws + o; o += (size_t)N * 4;
  float* pE     = ws + o; o += (size_t)Etot * 4;
  float* stats  = ws + o; o += 256;
  float* counts = ws + o; o += 64;
  (void)ws_size; (void)n_in; (void)out_size;

  const int B = 256;

  // ---------- layer 1: 5 -> 4x32 (concat 128), BN, ELU ----------
  k_gemm_in5<<<cdiv((long)N * 128, B), B, 0, stream>>>(x, W1, bufHW, N);
  k_attn<<<cdiv((long)N * 4, B), B, 0, stream>>>(bufHW, a1s, a1d, aS, aD, N, 4, 2, 32);
  run_edges(stream, ei, Eorig, Etot, N, bufHW, aS, aD, mM, dn, pE, bufAgg,
            /*Hh=*/4, /*hShift=*/2, /*CH=*/128, /*cShift=*/5, /*tpeShift=*/5);
  k_fill<<<1, 256, 0, stream>>>(stats, 0.f, 256);
  k_bn_stats<<<256, 128, 0, stream>>>(bufAgg, b1, stats, N, 128);
  k_bn_elu<<<cdiv((long)N * 128, B), B, 0, stream>>>(bufAgg, b1, stats, g1, be1, bufH, N, 128);

  // ---------- layer 2: 128 -> 4x32 (concat 128), BN, ELU ----------
  k_gemm_wmma<128><<<cdiv(N, 16), 256, 0, stream>>>(bufH, W2, bufHW, N);
  k_attn<<<cdiv((long)N * 4, B), B, 0, stream>>>(bufHW, a2s, a2d, aS, aD, N, 4, 2, 32);
  run_edges(stream, ei, Eorig, Etot, N, bufHW, aS, aD, mM, dn, pE, bufAgg,
            4, 2, 128, 5, 5);
  k_fill<<<1, 256, 0, stream>>>(stats, 0.f, 256);
  k_bn_stats<<<256, 128, 0, stream>>>(bufAgg, b2, stats, N, 128);
  k_bn_elu<<<cdiv((long)N * 128, B), B, 0, stream>>>(bufAgg, b2, stats, g2, be2, bufH, N, 128);

  // ---------- layer 3: 128 -> 1x64 (concat=False == identity for 1 head), ELU ----------
  k_gemm_wmma<64><<<cdiv(N, 16), 128, 0, stream>>>(bufH, W3, bufHW, N);
  k_attn<<<cdiv((long)N, B), B, 0, stream>>>(bufHW, a3s, a3d, aS, aD, N, 1, 0, 64);
  run_edges(stream, ei, Eorig, Etot, N, bufHW, aS, aD, mM, dn, pE, bufAgg,
            /*Hh=*/1, /*hShift=*/0, /*CH=*/64, /*cShift=*/6, /*tpeShift=*/4);
  k_elu_bias<<<cdiv((long)N * 64, B), B, 0, stream>>>(bufAgg, b3, bufH, (long)N * 64, 64);

  // ---------- pooling: [G, mean(64) || max(64)] ----------
  k_pool_init<<<cdiv(G * 128, B), B, 0, stream>>>(out, counts, G);
  k_pool_count<<<cdiv(N, B), B, 0, stream>>>(batch, counts, N);
  k_pool_accum<<<cdiv((long)N * 64, B), B, 0, stream>>>(bufH, batch, out, N);
  k_pool_final<<<cdiv(G * 128, B), B, 0, stream>>>(out, counts, G);
}